// DynamicDWConv_36575941493274
// MI455X (gfx1250) — compile-verified
//
#include <hip/hip_runtime.h>

typedef float v2f __attribute__((ext_vector_type(2)));
typedef float v4f __attribute__((ext_vector_type(4)));
typedef float v8f __attribute__((ext_vector_type(8)));

constexpr int Bn   = 16;     // batch
constexpr int Cc   = 96;     // channels
constexpr int Rr   = 24;     // reduced channels (C/4)
constexpr int Hh   = 128;
constexpr int Ww   = 128;
constexpr int HW   = Hh * Ww;        // 16384
constexpr int CKK  = Cc * 9;         // 864 = 54 * 16
constexpr float BN_EPS = 1e-5f;

// ---------------------------------------------------------------------------
// Kernel 1: global average pool over H*W for each (b,c) plane.
// 1536 blocks x 256 threads; each thread sums 64 floats via 16 x b128 loads.
// ---------------------------------------------------------------------------
__global__ __launch_bounds__(256) void pool_kernel(const float* __restrict__ x,
                                                   float* __restrict__ pooled) {
    const int plane = blockIdx.x;
    const float* p = x + (size_t)plane * HW;
    const int tid = threadIdx.x;
    float s = 0.f;
#pragma unroll
    for (int j = 0; j < 16; ++j) {
        v4f v = *(const v4f*)(p + j * 1024 + tid * 4);   // coalesced b128
        s += v.x + v.y + v.z + v.w;
    }
    __shared__ float red[256];
    red[tid] = s;
    __syncthreads();
    for (int off = 128; off > 0; off >>= 1) {
        if (tid < off) red[tid] += red[tid + off];
        __syncthreads();
    }
    if (tid == 0) pooled[plane] = red[0] * (1.f / 16384.f);
}

// ---------------------------------------------------------------------------
// Kernel 2: weight generation with f32 WMMA (V_WMMA_F32_16X16X4_F32).
//   GEMM1: y(16x24) = pooled(16x96) @ w1.T  -> BN(eval) -> ReLU   (waves 0,1)
//   GEMM2: dw(16x864) = y @ w2.T + b2                             (8 waves)
// M = batch = 16 exactly matches the WMMA tile. One 256-thread block.
// pooled and a zero-padded w1 are staged in LDS so every fragment gather is
// branch-free (no exec-mask churn around the WMMA chain).
// ---------------------------------------------------------------------------
__global__ __launch_bounds__(256) void weightgen_kernel(
    const float* __restrict__ pooled,   // (16,96)
    const float* __restrict__ w1,       // (24,96)
    const float* __restrict__ gamma, const float* __restrict__ beta,
    const float* __restrict__ rmean, const float* __restrict__ rvar,
    const float* __restrict__ w2,       // (864,24)
    const float* __restrict__ b2,       // (864,)
    float* __restrict__ dw)             // (16,864)
{
    __shared__ float pls[16 * 96];      // pooled staged
    __shared__ float w1ls[32 * 96];     // w1 padded: rows 24..31 are zero
    __shared__ float yls[16 * 24];      // intermediate y, row-major (b, r)

    const int tid = threadIdx.x;
    // stage pooled: 384 float4, fully coalesced
    for (int i = tid; i < 384; i += 256)
        *(v4f*)(pls + i * 4) = *(const v4f*)(pooled + i * 4);
    // stage w1 with zero padding: 32 rows x 24 float4
    for (int i = tid; i < 768; i += 256) {
        const int row = i / 24, q = i % 24;
        v4f v = {};
        if (row < Rr) v = *(const v4f*)(w1 + row * 96 + q * 4);
        *(v4f*)(w1ls + row * 96 + q * 4) = v;
    }
    __syncthreads();

    const int lane = tid & 31;
    const int wave = tid >> 5;
    const int l16  = lane & 15;               // M for A-frag, N for B-frag
    const int koff = (lane >> 4) << 1;        // half-wave K offset: 0 or 2
    const int mhi  = (lane < 16) ? 0 : 8;     // C/D row offset per half-wave

    // ---- GEMM1 + BN + ReLU: two 16x16 N-tiles (r = 0..15, 16..31-padded) ----
    if (wave < 2) {
        const int n = wave * 16 + l16;        // 0..31; rows >=24 read zeros
        v8f acc = {};
#pragma unroll
        for (int k0 = 0; k0 < 96; k0 += 4) {
            v2f a, b;
            a.x = pls[l16 * 96 + k0 + koff];          // A: M=l16, K=k0+koff
            a.y = pls[l16 * 96 + k0 + koff + 1];
            b.x = w1ls[n * 96 + k0 + koff];           // B[k][n] = w1[n][k]
            b.y = w1ls[n * 96 + k0 + koff + 1];
            acc = __builtin_amdgcn_wmma_f32_16x16x4_f32(
                      false, a, false, b, (short)0, acc, false, false);
        }
        const int nc = (n < Rr) ? n : (Rr - 1);       // clamp: branch-free loads
        const float sc = gamma[nc] * rsqrtf(rvar[nc] + BN_EPS);
        const float sh = beta[nc] - rmean[nc] * sc;
        if (n < Rr) {
#pragma unroll
            for (int i = 0; i < 8; ++i)
                yls[(i + mhi) * 24 + n] = fmaxf(acc[i] * sc + sh, 0.f);
        }
    }
    __syncthreads();

    // ---- GEMM2: 54 N-tiles of 16 over 864 outputs, K=24 (6 WMMA steps) ----
    float ax[6], ay[6];
#pragma unroll
    for (int kk = 0; kk < 6; ++kk) {          // A-fragments reused across tiles
        ax[kk] = yls[l16 * 24 + kk * 4 + koff];
        ay[kk] = yls[l16 * 24 + kk * 4 + koff + 1];
    }
    for (int t = wave; t < 54; t += 8) {
        const int o = t * 16 + l16;           // output index in [0, 864)
        const float* w2o = w2 + o * 24;       // B[k][n] = w2[o][k]
        v8f acc = {};
#pragma unroll
        for (int kk = 0; kk < 6; ++kk) {
            v2f a, b;
            a.x = ax[kk]; a.y = ay[kk];
            b.x = w2o[kk * 4 + koff];
            b.y = w2o[kk * 4 + koff + 1];
            acc = __builtin_amdgcn_wmma_f32_16x16x4_f32(
                      false, a, false, b, (short)0, acc, false, false);
        }
        const float bb = b2[o];
#pragma unroll
        for (int i = 0; i < 8; ++i)
            dw[(i + mhi) * CKK + o] = acc[i] + bb;
    }
}

// ---------------------------------------------------------------------------
// Kernel 3: depthwise 3x3 conv, pad 1, per-(b,c) dynamic weights + bias.
// Bandwidth-bound (~300 MB total traffic -> ~13 us at 23.3 TB/s).
// Per block: one plane x 32-row tile. LDS tile 34 x 136 floats: data at col 4
// (16B aligned for b128 ds stores), zero halo columns at 3 and 132 so the
// inner compute loop is branch-free. Each thread: 4 rows x 4 cols (float4 out).
// ---------------------------------------------------------------------------
constexpr int TILE_H  = 32;
constexpr int TROWS   = TILE_H + 2;  // 34
constexpr int TSTRIDE = 136;         // 128 data + halos + alignment pad

__global__ __launch_bounds__(256) void dwconv_kernel(
    const float* __restrict__ x, const float* __restrict__ dw,
    const float* __restrict__ bias, float* __restrict__ out)
{
    __shared__ float tile[TROWS * TSTRIDE];
    const int plane = blockIdx.x;             // b*96 + c
    const int y0 = blockIdx.y * TILE_H;
    const int tid = threadIdx.x;
    const float* xp = x + (size_t)plane * HW;

    if (tid < 2 * TROWS) {                    // zero halo columns
        const int r = tid >> 1;
        tile[r * TSTRIDE + ((tid & 1) ? 132 : 3)] = 0.f;
    }
    for (int i = tid; i < TROWS * 32; i += 256) {   // 34 rows x 32 float4
        const int r = i >> 5, q = i & 31;
        const int gy = y0 - 1 + r;
        v4f v = {};
        if (gy >= 0 && gy < Hh) v = *(const v4f*)(xp + gy * Ww + q * 4);
        *(v4f*)(tile + r * TSTRIDE + 4 + q * 4) = v;  // 16B-aligned ds b128
    }
    __syncthreads();

    const float* wk = dw + plane * 9;         // block-uniform -> scalar loads
    const float w00 = wk[0], w01 = wk[1], w02 = wk[2];
    const float w10 = wk[3], w11 = wk[4], w12 = wk[5];
    const float w20 = wk[6], w21 = wk[7], w22 = wk[8];
    const float bc = bias[plane % Cc];

    const int tx = tid & 31, ty = tid >> 5;
    const int x4 = tx * 4;
    float* op = out + (size_t)plane * HW;
#pragma unroll
    for (int ry = 0; ry < 4; ++ry) {
        const int yl = ty + ry * 8;           // local output row 0..31
        const float* r0 = tile + yl * TSTRIDE + 4 + x4;   // input row y-1
        const float* r1 = r0 + TSTRIDE;
        const float* r2 = r1 + TSTRIDE;
        v4f m0 = *(const v4f*)r0; const float L0 = r0[-1], E0 = r0[4];
        v4f m1 = *(const v4f*)r1; const float L1 = r1[-1], E1 = r1[4];
        v4f m2 = *(const v4f*)r2; const float L2 = r2[-1], E2 = r2[4];
        v4f o;
        o.x = w00*L0   + w01*m0.x + w02*m0.y
            + w10*L1   + w11*m1.x + w12*m1.y
            + w20*L2   + w21*m2.x + w22*m2.y + bc;
        o.y = w00*m0.x + w01*m0.y + w02*m0.z
            + w10*m1.x + w11*m1.y + w12*m1.z
            + w20*m2.x + w21*m2.y + w22*m2.z + bc;
        o.z = w00*m0.y + w01*m0.z + w02*m0.w
            + w10*m1.y + w11*m1.z + w12*m1.w
            + w20*m2.y + w21*m2.z + w22*m2.w + bc;
        o.w = w00*m0.z + w01*m0.w + w02*E0
            + w10*m1.z + w11*m1.w + w12*E1
            + w20*m2.z + w21*m2.w + w22*E2 + bc;
        *(v4f*)(op + (y0 + yl) * Ww + x4) = o;           // b128 store
    }
}

// ---------------------------------------------------------------------------
extern "C" void kernel_launch(void* const* d_in, const int* in_sizes, int n_in,
                              void* d_out, int out_size, void* d_ws, size_t ws_size,
                              hipStream_t stream) {
    const float* x_att = (const float*)d_in[0];
    const float* x_fea = (const float*)d_in[1];
    const float* w1    = (const float*)d_in[2];
    const float* gamma = (const float*)d_in[3];
    const float* beta  = (const float*)d_in[4];
    const float* rmean = (const float*)d_in[5];
    const float* rvar  = (const float*)d_in[6];
    const float* w2    = (const float*)d_in[7];
    const float* b2    = (const float*)d_in[8];
    const float* bias  = (const float*)d_in[9];
    float* out = (float*)d_out;

    float* pooled = (float*)d_ws;                       // 1536 floats
    float* dwbuf  = (float*)((char*)d_ws + 8192);       // 16*864 floats

    pool_kernel<<<dim3(Bn * Cc), dim3(256), 0, stream>>>(x_att, pooled);
    weightgen_kernel<<<dim3(1), dim3(256), 0, stream>>>(
        pooled, w1, gamma, beta, rmean, rvar, w2, b2, dwbuf);
    dwconv_kernel<<<dim3(Bn * Cc, Hh / TILE_H), dim3(256), 0, stream>>>(
        x_fea, dwbuf, bias, out);
}